// GATValueNetwork_81638738363132
// MI455X (gfx1250) — compile-verified
//
#include <hip/hip_runtime.h>
#include <math.h>

typedef __attribute__((ext_vector_type(2))) float v2f;
typedef __attribute__((ext_vector_type(8))) float v8f;

#define HEADS 4
#define CCH 64
#define HIDF 256
#define NEG_SLOPE 0.2f
#define LN_EPS 1e-5f

__device__ __forceinline__ float eluf(float x) { return x > 0.f ? x : (expf(x) - 1.f); }
__device__ __forceinline__ float leakyf(float x) { return x > 0.f ? x : NEG_SLOPE * x; }

// monotone float -> uint key (for atomicMax-based segment max)
__device__ __forceinline__ unsigned fkey(float f) {
    unsigned u = __float_as_uint(f);
    return (u & 0x80000000u) ? ~u : (u | 0x80000000u);
}
__device__ __forceinline__ float funkey(unsigned k) {
    unsigned u = (k & 0x80000000u) ? (k ^ 0x80000000u) : ~k;
    return __uint_as_float(u);
}

__device__ __forceinline__ float waveRedSum(float v) {
    for (int off = 16; off > 0; off >>= 1) v += __shfl_xor(v, off, 32);
    return v;
}

// ---------------------------------------------------------------------------
// Generic fp32 WMMA GEMM: C[M x Nout] = A[M x K] (row stride lda) * W^T,
// W is [Nout x K] row-major. One wave per 16x16 output tile, K-loop of
// V_WMMA_F32_16X16X4_F32. Optional bias + ELU epilogue (projection layer).
// M, Nout multiples of 16; K multiple of 4.
// ---------------------------------------------------------------------------
__global__ void gemm_wmma_f32(const float* __restrict__ A, int lda,
                              const float* __restrict__ W,
                              float* __restrict__ C,
                              int M, int K, int Nout,
                              const float* __restrict__ bias, int elu_act) {
    int wid  = (blockIdx.x * blockDim.x + threadIdx.x) >> 5;
    int lane = threadIdx.x & 31;
    int tilesN = Nout >> 4;
    int totalTiles = (M >> 4) * tilesN;
    if (wid >= totalTiles) return;           // wave-uniform guard, EXEC stays all-1
    int tm = wid / tilesN;
    int tn = wid - tm * tilesN;
    int m  = (tm << 4) + (lane & 15);
    int n  = (tn << 4) + (lane & 15);
    int kb = (lane >> 4) << 1;
    const float* arow = A + (size_t)m * lda;
    const float* brow = W + (size_t)n * K;
    v8f acc = {};
    for (int k = 0; k < K; k += 4) {
        v2f a, b;
        a.x = arow[k + kb];
        a.y = arow[k + kb + 1];
        b.x = brow[k + kb];
        b.y = brow[k + kb + 1];
        acc = __builtin_amdgcn_wmma_f32_16x16x4_f32(false, a, false, b,
                                                    (short)0, acc, false, false);
    }
    int rbase = (tm << 4) + ((lane >> 4) << 3);
    int col   = (tn << 4) + (lane & 15);
    float bv  = bias ? bias[col] : 0.f;
    for (int i = 0; i < 8; ++i) {
        float v = acc[i] + bv;
        if (elu_act) v = eluf(v);
        C[(size_t)(rbase + i) * Nout + col] = v;
    }
}

// asrc[n,h] = <h[n,h,:], a_s[h,:]>, adst likewise. One thread per (n,h).
__global__ void attn_coef(const float* __restrict__ hmat,
                          const float* __restrict__ a_s,
                          const float* __restrict__ a_d,
                          float* __restrict__ asrc, float* __restrict__ adst, int NH) {
    int idx = blockIdx.x * blockDim.x + threadIdx.x;
    if (idx >= NH) return;
    int n = idx >> 2, h = idx & 3;
    const float* row = hmat + (size_t)n * HIDF + h * CCH;
    const float* ws  = a_s + h * CCH;
    const float* wd  = a_d + h * CCH;
    float ss = 0.f, sd = 0.f;
    for (int c = 0; c < CCH; ++c) { float v = row[c]; ss += v * ws[c]; sd += v * wd[c]; }
    asrc[idx] = ss;
    adst[idx] = sd;
}

// init segment-max keys with the self-loop alpha
__global__ void softmax_init(const float* __restrict__ asrc, const float* __restrict__ adst,
                             unsigned* __restrict__ amaxk, int NH) {
    int idx = blockIdx.x * blockDim.x + threadIdx.x;
    if (idx >= NH) return;
    amaxk[idx] = fkey(leakyf(asrc[idx] + adst[idx]));
}

// segment max over edges via u32 atomicMax on monotone keys
__global__ void edge_max(const int* __restrict__ src, const int* __restrict__ dst,
                         const float* __restrict__ asrc, const float* __restrict__ adst,
                         unsigned* __restrict__ amaxk, int EH) {
    int idx = blockIdx.x * blockDim.x + threadIdx.x;
    if (idx >= EH) return;
    int e = idx >> 2, h = idx & 3;
    int s = src[e], d = dst[e];
    float a = leakyf(asrc[s * HEADS + h] + adst[d * HEADS + h]);
    atomicMax(&amaxk[d * HEADS + h], fkey(a));
}

// self-loop contribution: one wave per node; channel c = lane + 32*i maps to
// head i>>1 (compile-time after unroll) -> coalesced 128B stores.
__global__ void acc_init(const float* __restrict__ hmat,
                         const float* __restrict__ asrc, const float* __restrict__ adst,
                         const unsigned* __restrict__ amaxk,
                         float* __restrict__ denom, float* __restrict__ hacc, int N) {
    int wid  = (blockIdx.x * blockDim.x + threadIdx.x) >> 5;
    int lane = threadIdx.x & 31;
    if (wid >= N) return;
    int n = wid;
    float ex[HEADS];
#pragma unroll
    for (int h = 0; h < HEADS; ++h) {
        float aself = leakyf(asrc[n * HEADS + h] + adst[n * HEADS + h]);
        ex[h] = expf(aself - funkey(amaxk[n * HEADS + h]));
    }
    if (lane == 0) {
#pragma unroll
        for (int h = 0; h < HEADS; ++h) denom[n * HEADS + h] = ex[h];
    }
    const float* row = hmat + (size_t)n * HIDF;
    float* orow      = hacc + (size_t)n * HIDF;
#pragma unroll
    for (int i = 0; i < 8; ++i) {
        int c = lane + 32 * i;
        orow[c] = row[c] * ex[i >> 1];
    }
}

// heavy edge stage: one wave per edge covering ALL 4 heads.
// ex[] lives in registers with static indices (head = i>>1 after unroll);
// lane 0 posts the 4 denom atomics, all lanes do 8 coalesced loads + atomics.
__global__ void edge_accum(const int* __restrict__ src, const int* __restrict__ dst,
                           const float* __restrict__ asrc, const float* __restrict__ adst,
                           const unsigned* __restrict__ amaxk,
                           const float* __restrict__ hmat,
                           float* __restrict__ denom, float* __restrict__ hacc, int E) {
    int wid  = (blockIdx.x * blockDim.x + threadIdx.x) >> 5;
    int lane = threadIdx.x & 31;
    if (wid >= E) return;
    int e = wid;
    int s = src[e], d = dst[e];
    float ex[HEADS];
#pragma unroll
    for (int h = 0; h < HEADS; ++h) {
        float a = leakyf(asrc[s * HEADS + h] + adst[d * HEADS + h]);
        ex[h] = expf(a - funkey(amaxk[d * HEADS + h]));
    }
    if (lane == 0) {
#pragma unroll
        for (int h = 0; h < HEADS; ++h) atomicAdd(&denom[d * HEADS + h], ex[h]);
    }
    const float* hrow = hmat + (size_t)s * HIDF;
    float* orow       = hacc + (size_t)d * HIDF;
#pragma unroll
    for (int i = 0; i < 8; ++i) {
        int c = lane + 32 * i;
        atomicAdd(&orow[c], ex[i >> 1] * hrow[c]);
    }
}

// concat layers: normalize by denom, +bias, LayerNorm(256), ELU. One wave/node.
__global__ void finalize_concat(const float* __restrict__ hacc, const float* __restrict__ denom,
                                const float* __restrict__ bias, const float* __restrict__ g,
                                const float* __restrict__ be, float* __restrict__ out, int N) {
    int wid  = (blockIdx.x * blockDim.x + threadIdx.x) >> 5;
    int lane = threadIdx.x & 31;
    if (wid >= N) return;
    size_t base = (size_t)wid * HIDF;
    float dn[HEADS];
#pragma unroll
    for (int h = 0; h < HEADS; ++h) dn[h] = 1.f / (denom[wid * HEADS + h] + 1e-16f);
    float v[8];
    float s = 0.f;
#pragma unroll
    for (int i = 0; i < 8; ++i) {
        int c = lane + 32 * i;
        v[i] = hacc[base + c] * dn[i >> 1] + bias[c];
        s += v[i];
    }
    float mu = waveRedSum(s) * (1.f / HIDF);
    float q = 0.f;
#pragma unroll
    for (int i = 0; i < 8; ++i) { float dlt = v[i] - mu; q += dlt * dlt; }
    float rstd = rsqrtf(waveRedSum(q) * (1.f / HIDF) + LN_EPS);
#pragma unroll
    for (int i = 0; i < 8; ++i) {
        int c = lane + 32 * i;
        float y = (v[i] - mu) * rstd * g[c] + be[c];
        out[base + c] = eluf(y);
    }
}

// last layer: mean over heads -> 64 feats, +bias, LayerNorm(64), ELU. One wave/node.
__global__ void finalize_mean(const float* __restrict__ hacc, const float* __restrict__ denom,
                              const float* __restrict__ bias, const float* __restrict__ g,
                              const float* __restrict__ be, float* __restrict__ out, int N) {
    int wid  = (blockIdx.x * blockDim.x + threadIdx.x) >> 5;
    int lane = threadIdx.x & 31;
    if (wid >= N) return;
    size_t base = (size_t)wid * HIDF;
    float dn[HEADS];
#pragma unroll
    for (int h = 0; h < HEADS; ++h) dn[h] = 1.f / (denom[wid * HEADS + h] + 1e-16f);
    float v[2];
    float s = 0.f;
#pragma unroll
    for (int i = 0; i < 2; ++i) {
        int c = lane + 32 * i;
        float acc = 0.f;
#pragma unroll
        for (int h = 0; h < HEADS; ++h) acc += hacc[base + h * CCH + c] * dn[h];
        v[i] = acc * (1.f / HEADS) + bias[c];
        s += v[i];
    }
    float mu = waveRedSum(s) * (1.f / CCH);
    float q = 0.f;
#pragma unroll
    for (int i = 0; i < 2; ++i) { float dlt = v[i] - mu; q += dlt * dlt; }
    float rstd = rsqrtf(waveRedSum(q) * (1.f / CCH) + LN_EPS);
#pragma unroll
    for (int i = 0; i < 2; ++i) {
        int c = lane + 32 * i;
        float y = (v[i] - mu) * rstd * g[c] + be[c];
        out[(size_t)wid * CCH + c] = eluf(y);
    }
}

// value head: lane j computes hidden_j (32 hidden), then 4 outputs by shuffle reduce.
__global__ void value_head(const float* __restrict__ hin,
                           const float* __restrict__ vw1, const float* __restrict__ vb1,
                           const float* __restrict__ vw2, const float* __restrict__ vb2,
                           float* __restrict__ out, int N) {
    int wid  = (blockIdx.x * blockDim.x + threadIdx.x) >> 5;
    int lane = threadIdx.x & 31;
    if (wid >= N) return;
    const float* row  = hin + (size_t)wid * CCH;
    const float* wrow = vw1 + lane * CCH;
    float hid = vb1[lane];
    for (int c = 0; c < CCH; ++c) hid += row[c] * wrow[c];
    hid = eluf(hid);
#pragma unroll
    for (int k = 0; k < 4; ++k) {
        float p = vw2[k * 32 + lane] * hid;
        p = waveRedSum(p);
        if (lane == 0) out[wid * 4 + k] = p + vb2[k];
    }
}

extern "C" void kernel_launch(void* const* d_in, const int* in_sizes, int n_in,
                              void* d_out, int out_size, void* d_ws, size_t ws_size,
                              hipStream_t stream) {
    const float* x      = (const float*)d_in[0];
    const int*   esrc   = (const int*)d_in[1];
    const int*   edst   = (const int*)d_in[2];
    const float* proj_w = (const float*)d_in[3];
    const float* proj_b = (const float*)d_in[4];
    const float* W[3]  = {(const float*)d_in[5],  (const float*)d_in[11], (const float*)d_in[17]};
    const float* AS[3] = {(const float*)d_in[6],  (const float*)d_in[12], (const float*)d_in[18]};
    const float* AD[3] = {(const float*)d_in[7],  (const float*)d_in[13], (const float*)d_in[19]};
    const float* B[3]  = {(const float*)d_in[8],  (const float*)d_in[14], (const float*)d_in[20]};
    const float* G[3]  = {(const float*)d_in[9],  (const float*)d_in[15], (const float*)d_in[21]};
    const float* BE[3] = {(const float*)d_in[10], (const float*)d_in[16], (const float*)d_in[22]};
    const float* vw1 = (const float*)d_in[23];
    const float* vb1 = (const float*)d_in[24];
    const float* vw2 = (const float*)d_in[25];
    const float* vb2 = (const float*)d_in[26];

    const int N = in_sizes[0] / 128;   // 50000 (multiple of 16)
    const int E = in_sizes[1];         // 800000
    const int NH = N * HEADS;
    const int EH = E * HEADS;

    float* ws     = (float*)d_ws;
    float* hin    = ws;                          // N*256
    float* hmat   = hin  + (size_t)N * HIDF;     // N*256
    float* hacc   = hmat + (size_t)N * HIDF;     // N*256
    float* asrc   = hacc + (size_t)N * HIDF;     // N*4
    float* adst   = asrc + (size_t)NH;           // N*4
    unsigned* amk = (unsigned*)(adst + (size_t)NH);  // N*4
    float* denom  = (float*)(amk + (size_t)NH);      // N*4

    dim3 blk(256);

    // input projection: hin[N,64] = ELU(x @ proj_w^T + proj_b)
    {
        int tiles = (N / 16) * (64 / 16);
        gemm_wmma_f32<<<(tiles * 32 + 255) / 256, blk, 0, stream>>>(
            x, 128, proj_w, hin, N, 128, 64, proj_b, 1);
    }

    int in_ld = 64;
    for (int l = 0; l < 3; ++l) {
        int K = (l == 0) ? 64 : 256;
        int tiles = (N / 16) * (HIDF / 16);
        gemm_wmma_f32<<<(tiles * 32 + 255) / 256, blk, 0, stream>>>(
            hin, in_ld, W[l], hmat, N, K, HIDF, nullptr, 0);
        attn_coef<<<(NH + 255) / 256, blk, 0, stream>>>(hmat, AS[l], AD[l], asrc, adst, NH);
        softmax_init<<<(NH + 255) / 256, blk, 0, stream>>>(asrc, adst, amk, NH);
        edge_max<<<(EH + 255) / 256, blk, 0, stream>>>(esrc, edst, asrc, adst, amk, EH);
        acc_init<<<(N * 32 + 255) / 256, blk, 0, stream>>>(hmat, asrc, adst, amk, denom, hacc, N);
        edge_accum<<<((size_t)E * 32 + 255) / 256, blk, 0, stream>>>(
            esrc, edst, asrc, adst, amk, hmat, denom, hacc, E);
        if (l < 2) {
            finalize_concat<<<(N * 32 + 255) / 256, blk, 0, stream>>>(
                hacc, denom, B[l], G[l], BE[l], hin, N);
            in_ld = HIDF;
        } else {
            finalize_mean<<<(N * 32 + 255) / 256, blk, 0, stream>>>(
                hacc, denom, B[l], G[l], BE[l], hin, N);
        }
    }

    value_head<<<(N * 32 + 255) / 256, blk, 0, stream>>>(
        hin, vw1, vb1, vw2, vb2, (float*)d_out, N);
}